// ProductionMoELayer_34007551050354
// MI455X (gfx1250) — compile-verified
//
#include <hip/hip_runtime.h>
#include <math.h>

// ---------------- problem constants ----------------
#define N_TOK   8192          // B*S = 4*2048
#define EMBED   1024
#define FF      4096
#define NEXP    8
#define TOPK    2
#define CAP     1280          // int(8192/8 * 1.25)
#define ROWS    (TOPK*CAP)    // 2560 combined rows per expert (both k slots)
#define LB_W    0.01f
#define Z_W     0.001f

// ---------------- WMMA / TDM types (gfx1250, wave32) ----------------
typedef float v2f  __attribute__((ext_vector_type(2)));
typedef float v8f  __attribute__((ext_vector_type(8)));
typedef unsigned int u32x4 __attribute__((ext_vector_type(4)));
typedef int   i32x4 __attribute__((ext_vector_type(4)));
typedef int   i32x8 __attribute__((ext_vector_type(8)));

// ---------------- tiling ----------------
#define KC 32     // K-chunk staged in LDS (8 wmma k-steps of 4)
#define TM 64     // rows per workgroup tile (4 WMMA M-tiles)
#define TN 128    // cols per workgroup tile (8 WMMA N-tiles, 4 per wave-column)

// ---------------- Tensor Data Mover support ----------------
#if defined(__has_builtin)
# if __has_builtin(__builtin_amdgcn_tensor_load_to_lds) && \
     __has_builtin(__builtin_amdgcn_s_wait_tensorcnt)
#  define HAVE_TDM 1
# endif
#endif
#ifndef HAVE_TDM
# define HAVE_TDM 0
#endif

// Truncate a generic pointer to the LDS byte offset (aperture: addr[31:0]).
__device__ __forceinline__ unsigned lds_off(const void* p) {
  return (unsigned)(size_t)p;
}

#if HAVE_TDM
// Issue a 2D TDM tile load: tile_y rows x tile_x fp32 elements, row stride
// `stride_elems` (fp32 units), from `gaddr` into LDS at byte offset `lds`.
// D# layout per CDNA5 ISA. One issue per wave.
__device__ __forceinline__ void tdm_issue(unsigned lds, const void* gaddr,
                                          unsigned tile_x, unsigned tile_y,
                                          unsigned long long stride_elems) {
  const unsigned long long ga = (unsigned long long)(size_t)gaddr;
  u32x4 g0 = { 1u,                                   // count=1, no gather
               lds,                                  // lds_addr
               (unsigned)(ga & 0xffffffffu),         // global_addr[31:0]
               (unsigned)((ga >> 32) & 0x01ffffffu)  // global_addr[56:32]
                 | (2u << 30) };                     // type=2 ("image")
  i32x8 g1 = {
      (int)(2u << 16),                               // data_size=2 -> 4 bytes
      (int)((tile_x & 0xffffu) << 16),               // tensor_dim0[15:0]
      (int)(((tile_x >> 16) & 0xffffu) |             // tensor_dim0[31:16]
            ((tile_y & 0xffffu) << 16)),             // tensor_dim1[15:0]
      (int)(((tile_y >> 16) & 0xffffu) |             // tensor_dim1[31:16]
            ((tile_x & 0xffffu) << 16)),             // tile_dim0
      (int)(tile_y & 0xffffu),                       // tile_dim1 (tile_dim2=0)
      (int)(stride_elems & 0xffffffffu),             // dim0_stride[31:0]
      (int)((stride_elems >> 32) & 0xffffu),         // dim0_stride[47:32]
      0 };
  i32x4 gz = {0, 0, 0, 0};
#if defined(__clang_major__) && __clang_major__ >= 23
  i32x8 gz8 = {0, 0, 0, 0, 0, 0, 0, 0};
  __builtin_amdgcn_tensor_load_to_lds(g0, g1, gz, gz, gz8, 0);
#else
  __builtin_amdgcn_tensor_load_to_lds(g0, g1, gz, gz, 0);
#endif
}
#endif

// Stage a dense rows x KC fp32 tile into LDS. TDM path: wave 0 issues one DMA
// per tile; fallback: cooperative vector loads.
__device__ __forceinline__ void stage_tile(float* dst, const float* g,
                                           int rows, long long stride,
                                           int tid, int wave) {
#if HAVE_TDM
  (void)tid;
  if (wave == 0)
    tdm_issue(lds_off(dst), g, (unsigned)KC, (unsigned)rows,
              (unsigned long long)stride);
#else
  (void)wave;
  for (int i = tid; i < rows * KC / 4; i += 256) {
    const int r = i / (KC / 4), c4 = i % (KC / 4);
    ((float4*)dst)[i] = *(const float4*)&g[(size_t)r * stride + c4 * 4];
  }
#endif
}

__device__ __forceinline__ void stage_wait(int wave) {
#if HAVE_TDM
  if (wave == 0) __builtin_amdgcn_s_wait_tensorcnt(0);
#else
  (void)wave;
#endif
}

// =================================================================
// zero-fill helper
// =================================================================
__global__ void zero_kernel(float* __restrict__ p, size_t n) {
  size_t i = (size_t)blockIdx.x * blockDim.x + threadIdx.x;
  size_t stride = (size_t)gridDim.x * blockDim.x;
  for (; i < n; i += stride) p[i] = 0.0f;
}

// =================================================================
// gating: logits -> softmax -> top2 -> normalized weights + loss accums
// one wave per token; accum: [0]=sum lse^2, [1..8]=density, [9..16]=prob sums
// =================================================================
__global__ void gate_kernel(const float* __restrict__ x,
                            const float* __restrict__ gw,
                            int*   __restrict__ topi,
                            float* __restrict__ topv,
                            float* __restrict__ accum) {
  __shared__ float gwl[NEXP * EMBED];     // 32 KB gate weights staged once
  __shared__ float bacc[1 + 2 * NEXP];
  const int tid = threadIdx.x;
  for (int i = tid; i < NEXP * EMBED / 4; i += blockDim.x)
    ((float4*)gwl)[i] = ((const float4*)gw)[i];
  if (tid < 1 + 2 * NEXP) bacc[tid] = 0.0f;
  __syncthreads();

  const int lane = tid & 31, wave = tid >> 5;
  const int token = blockIdx.x * 8 + wave;
  const float* xr = x + (size_t)token * EMBED;

  float acc[NEXP];
#pragma unroll
  for (int e = 0; e < NEXP; ++e) acc[e] = 0.0f;
  for (int d = lane; d < EMBED; d += 32) {
    const float xv = xr[d];
#pragma unroll
    for (int e = 0; e < NEXP; ++e) acc[e] += xv * gwl[e * EMBED + d];
  }
#pragma unroll
  for (int e = 0; e < NEXP; ++e)
    for (int off = 16; off; off >>= 1) acc[e] += __shfl_down(acc[e], off, 32);

  if (lane == 0) {
    float m = acc[0];
#pragma unroll
    for (int e = 1; e < NEXP; ++e) m = fmaxf(m, acc[e]);
    float p[NEXP], s = 0.0f;
#pragma unroll
    for (int e = 0; e < NEXP; ++e) { p[e] = __expf(acc[e] - m); s += p[e]; }
    const float inv = 1.0f / s;
#pragma unroll
    for (int e = 0; e < NEXP; ++e) p[e] *= inv;
    const float lse = m + __logf(s);

    // top-2, lowest-index tie-break (matches jax.lax.top_k)
    int i0 = 0;
#pragma unroll
    for (int e = 1; e < NEXP; ++e) if (p[e] > p[i0]) i0 = e;
    int i1 = (i0 == 0) ? 1 : 0;
#pragma unroll
    for (int e = 0; e < NEXP; ++e) if (e != i0 && p[e] > p[i1]) i1 = e;
    const float v0 = p[i0], v1 = p[i1], vs = v0 + v1;

    topi[token * 2 + 0] = i0;
    topi[token * 2 + 1] = i1;
    topv[token * 2 + 0] = v0 / vs;
    topv[token * 2 + 1] = v1 / vs;

    atomicAdd(&bacc[0], lse * lse);
    atomicAdd(&bacc[1 + i0], 1.0f);
    atomicAdd(&bacc[1 + i1], 1.0f);
#pragma unroll
    for (int e = 0; e < NEXP; ++e) atomicAdd(&bacc[1 + NEXP + e], p[e]);
  }
  __syncthreads();
  if (tid < 1 + 2 * NEXP) atomicAdd(&accum[tid], bacc[tid]);
}

// =================================================================
// scan: token-order exclusive prefix per (k,expert) -> capacity slots.
// =================================================================
#define SCAN_T 1024
#define TPT    (N_TOK / SCAN_T)   // 8 tokens per thread
__global__ void scan_kernel(const int*   __restrict__ topi,
                            const float* __restrict__ topv,
                            int*   __restrict__ rowtok,
                            float* __restrict__ roww,
                            const float* __restrict__ accum,
                            float* __restrict__ loss_out) {
  __shared__ unsigned short cnt[TOPK * NEXP][SCAN_T];   // 32 KB
  const int tid = threadIdx.x;

  int local[TOPK * NEXP];
#pragma unroll
  for (int i = 0; i < TOPK * NEXP; ++i) local[i] = 0;
  const int base = tid * TPT;
  for (int i = 0; i < TPT; ++i) {
    const int t = base + i;
    local[0 * NEXP + topi[t * 2 + 0]]++;
    local[1 * NEXP + topi[t * 2 + 1]]++;
  }
#pragma unroll
  for (int i = 0; i < TOPK * NEXP; ++i) cnt[i][tid] = (unsigned short)local[i];
  __syncthreads();

  if (tid < TOPK * NEXP) {   // serial exclusive scan per (k,e) column
    unsigned int run = 0;
    for (int j = 0; j < SCAN_T; ++j) {
      const unsigned int c = cnt[tid][j];
      cnt[tid][j] = (unsigned short)run;
      run += c;
    }
  }
  __syncthreads();

  int off[TOPK * NEXP];
#pragma unroll
  for (int i = 0; i < TOPK * NEXP; ++i) off[i] = cnt[i][tid];
  for (int i = 0; i < TPT; ++i) {
    const int t = base + i;
#pragma unroll
    for (int k = 0; k < TOPK; ++k) {
      const int e   = topi[t * 2 + k];
      const int idx = k * NEXP + e;
      const int pos = off[idx]++;
      if (pos < CAP) {
        const int slot = e * ROWS + k * CAP + pos;
        rowtok[slot] = t;
        roww[slot]   = topv[t * 2 + k];
      }
    }
  }

  if (tid == 0) {
    const float z = accum[0] / (float)N_TOK * Z_W;
    float lb = 0.0f;
    for (int e = 0; e < NEXP; ++e) {
      const float dens = accum[1 + e] / (float)(N_TOK * TOPK);
      const float pm   = accum[1 + NEXP + e] / (float)N_TOK;
      lb += dens * pm;
    }
    loss_out[0] = z + (float)NEXP * lb * LB_W;
  }
}

// =================================================================
// gemm1: per expert, H[r,f] = silu(X Wg^T) * (X Wu^T)
// B tiles via TDM double-buffer; gathered A tile via register double-buffer.
// =================================================================
__global__ void __launch_bounds__(256)
gemm1_kernel(const float* __restrict__ x,
             const float* __restrict__ wig,
             const float* __restrict__ wiu,
             const int*   __restrict__ rowtok,
             float* __restrict__ H,
             int expert) {
  __shared__ float Al[2][TM * KC];     // 16 KB gathered X chunks
  __shared__ float Bg[2][TN * KC];     // 32 KB wi_gate chunks
  __shared__ float Bu[2][TN * KC];     // 32 KB wi_up chunks
  __shared__ int   tok[TM];

  const int tid  = threadIdx.x;
  const int lane = tid & 31, wave = tid >> 5;
  const int row0 = blockIdx.x * TM;
  const int f0   = blockIdx.y * TN;

  if (tid < TM) {
    const int t = rowtok[expert * ROWS + row0 + tid];
    tok[tid] = (t >= 0 && t < N_TOK) ? t : 0;   // clamp poison/stale slots
  }
  __syncthreads();

  const float* wg = wig + (size_t)expert * FF * EMBED + (size_t)f0 * EMBED;
  const float* wu = wiu + (size_t)expert * FF * EMBED + (size_t)f0 * EMBED;

  const int mt = wave & 3, ncol = wave >> 2;
  const int half = lane >> 4, l16 = lane & 15;

  v8f cg[4] = {};
  v8f cu[4] = {};

  // prologue: stage chunk 0 into buffer 0
  stage_tile(Bg[0], wg, TN, EMBED, tid, wave);
  stage_tile(Bu[0], wu, TN, EMBED, tid, wave);
  float4 areg[2];
#pragma unroll
  for (int t = 0; t < 2; ++t) {
    const int i = tid + t * 256, r = i / (KC / 4), c4 = i % (KC / 4);
    areg[t] = *(const float4*)&x[(size_t)tok[r] * EMBED + c4 * 4];
  }
#pragma unroll
  for (int t = 0; t < 2; ++t) ((float4*)Al[0])[tid + t * 256] = areg[t];
  stage_wait(wave);
  __syncthreads();

  int cur = 0;
  for (int k0 = 0; k0 < EMBED; k0 += KC) {
    const int nxt = cur ^ 1;
    const bool more = (k0 + KC) < EMBED;
    if (more) {
      // issue next chunk's DMA + gathered global loads before computing
      stage_tile(Bg[nxt], wg + (k0 + KC), TN, EMBED, tid, wave);
      stage_tile(Bu[nxt], wu + (k0 + KC), TN, EMBED, tid, wave);
#pragma unroll
      for (int t = 0; t < 2; ++t) {
        const int i = tid + t * 256, r = i / (KC / 4), c4 = i % (KC / 4);
        areg[t] = *(const float4*)&x[(size_t)tok[r] * EMBED + k0 + KC + c4 * 4];
      }
    }

#pragma unroll
    for (int kk = 0; kk < KC; kk += 4) {
      const v2f a = *(const v2f*)&Al[cur][(mt * 16 + l16) * KC + kk + 2 * half];
#pragma unroll
      for (int j = 0; j < 4; ++j) {
        const int nb = (ncol * 4 + j) * 16;
        const v2f bg = *(const v2f*)&Bg[cur][(nb + l16) * KC + kk + 2 * half];
        const v2f bu = *(const v2f*)&Bu[cur][(nb + l16) * KC + kk + 2 * half];
        cg[j] = __builtin_amdgcn_wmma_f32_16x16x4_f32(
            false, a, false, bg, (short)0, cg[j], false, false);
        cu[j] = __builtin_amdgcn_wmma_f32_16x16x4_f32(
            false, a, false, bu, (short)0, cu[j], false, false);
      }
    }

    if (more) {
#pragma unroll
      for (int t = 0; t < 2; ++t) ((float4*)Al[nxt])[tid + t * 256] = areg[t];
      stage_wait(wave);
    }
    __syncthreads();
    cur = nxt;
  }

  // epilogue: silu(gate) * up -> H
#pragma unroll
  for (int j = 0; j < 4; ++j) {
    const int col = f0 + (ncol * 4 + j) * 16 + l16;
#pragma unroll
    for (int v = 0; v < 8; ++v) {
      const int m = mt * 16 + v + 8 * half;
      const float g = cg[j][v], u = cu[j][v];
      const float hval = (g / (1.0f + __expf(-g))) * u;
      H[(size_t)(row0 + m) * FF + col] = hval;
    }
  }
}

// =================================================================
// gemm2: per expert, out_row = (H Wo^T) * slot_weight, scattered by token.
// Both A (H) and B (wo) tiles are dense -> full TDM double-buffer.
// =================================================================
__global__ void __launch_bounds__(256)
gemm2_kernel(const float* __restrict__ H,
             const float* __restrict__ wo,
             const int*   __restrict__ rowtok,
             const float* __restrict__ roww,
             float* __restrict__ out,
             int expert) {
  __shared__ float Al[2][TM * KC];     // 16 KB H chunks
  __shared__ float Bl[2][TN * KC];     // 32 KB wo chunks
  __shared__ int   tok[TM];
  __shared__ float wgt[TM];

  const int tid  = threadIdx.x;
  const int lane = tid & 31, wave = tid >> 5;
  const int row0 = blockIdx.x * TM;
  const int c0   = blockIdx.y * TN;

  if (tid < TM) {
    const int slot = expert * ROWS + row0 + tid;
    const int t = rowtok[slot];
    tok[tid] = (t >= 0 && t < N_TOK) ? t : 0;
    wgt[tid] = roww[slot];
  }
  __syncthreads();

  const float* ha = H + (size_t)row0 * FF;
  const float* wb = wo + (size_t)expert * EMBED * FF + (size_t)c0 * FF;

  const int mt = wave & 3, ncol = wave >> 2;
  const int half = lane >> 4, l16 = lane & 15;

  v8f c[4] = {};

  stage_tile(Al[0], ha, TM, FF, tid, wave);
  stage_tile(Bl[0], wb, TN, FF, tid, wave);
  stage_wait(wave);
  __syncthreads();

  int cur = 0;
  for (int k0 = 0; k0 < FF; k0 += KC) {
    const int nxt = cur ^ 1;
    const bool more = (k0 + KC) < FF;
    if (more) {
      stage_tile(Al[nxt], ha + (k0 + KC), TM, FF, tid, wave);
      stage_tile(Bl[nxt], wb + (k0 + KC), TN, FF, tid, wave);
    }

#pragma unroll
    for (int kk = 0; kk < KC; kk += 4) {
      const v2f a = *(const v2f*)&Al[cur][(mt * 16 + l16) * KC + kk + 2 * half];
#pragma unroll
      for (int j = 0; j < 4; ++j) {
        const int nb = (ncol * 4 + j) * 16;
        const v2f b = *(const v2f*)&Bl[cur][(nb + l16) * KC + kk + 2 * half];
        c[j] = __builtin_amdgcn_wmma_f32_16x16x4_f32(
            false, a, false, b, (short)0, c[j], false, false);
      }
    }

    if (more) stage_wait(wave);
    __syncthreads();
    cur = nxt;
  }

#pragma unroll
  for (int j = 0; j < 4; ++j) {
    const int col = c0 + (ncol * 4 + j) * 16 + l16;
#pragma unroll
    for (int v = 0; v < 8; ++v) {
      const int m = mt * 16 + v + 8 * half;
      const float w = wgt[m];
      if (w != 0.0f) {
        atomicAdd(&out[(size_t)tok[m] * EMBED + col], c[j][v] * w);
      }
    }
  }
}

// =================================================================
// host launcher
// =================================================================
extern "C" void kernel_launch(void* const* d_in, const int* in_sizes, int n_in,
                              void* d_out, int out_size, void* d_ws,
                              size_t ws_size, hipStream_t stream) {
  (void)in_sizes; (void)n_in; (void)out_size; (void)ws_size;

  const float* x      = (const float*)d_in[0];
  const float* gate_w = (const float*)d_in[1];
  const float* wig    = (const float*)d_in[2];
  const float* wiu    = (const float*)d_in[3];
  const float* wo     = (const float*)d_in[4];
  float* out = (float*)d_out;   // [N_TOK*EMBED] output + [1] aux loss

  // workspace layout (~288 KB tables + 40 MB H buffer)
  char* ws = (char*)d_ws;
  size_t off = 0;
  int*   topi   = (int*)(ws + off);   off += (size_t)N_TOK * 2 * sizeof(int);
  float* topv   = (float*)(ws + off); off += (size_t)N_TOK * 2 * sizeof(float);
  int*   rowtok = (int*)(ws + off);   off += (size_t)NEXP * ROWS * sizeof(int);
  float* roww   = (float*)(ws + off); off += (size_t)NEXP * ROWS * sizeof(float);
  float* accum  = (float*)(ws + off); off += 32 * sizeof(float);
  off = (off + 255) & ~(size_t)255;
  float* Hbuf   = (float*)(ws + off);  // ROWS * FF floats, reused per expert

  zero_kernel<<<256, 256, 0, stream>>>(out, (size_t)N_TOK * EMBED + 1);
  zero_kernel<<<32, 256, 0, stream>>>(roww, (size_t)NEXP * ROWS);
  zero_kernel<<<1, 32, 0, stream>>>(accum, 32);

  gate_kernel<<<N_TOK / 8, 256, 0, stream>>>(x, gate_w, topi, topv, accum);
  scan_kernel<<<1, SCAN_T, 0, stream>>>(topi, topv, rowtok, roww, accum,
                                        out + (size_t)N_TOK * EMBED);

  // experts sequential: each expert's 48 MB of weights stays hot in the
  // 192 MB L2 across its 1280-tile GEMM1 and 320-tile GEMM2.
  for (int e = 0; e < NEXP; ++e) {
    dim3 g1(ROWS / TM, FF / TN);      // 40 x 32
    gemm1_kernel<<<g1, 256, 0, stream>>>(x, wig, wiu, rowtok, Hbuf, e);
    dim3 g2(ROWS / TM, EMBED / TN);   // 40 x 8
    gemm2_kernel<<<g2, 256, 0, stream>>>(Hbuf, wo, rowtok, roww, out, e);
  }
}